// NodeSparsityLoss_60739427500434
// MI455X (gfx1250) — compile-verified
//
#include <hip/hip_runtime.h>

typedef __attribute__((ext_vector_type(2))) float v2f;
typedef __attribute__((ext_vector_type(8))) float v8f;

#define NPTS   4096
#define NTILES 256          // 4096 / 16
#define NBATCH 8
#define TARGET_F 0.2f

// Kernel 1: per-row nearest-neighbor squared distance via WMMA Gram tiles.
// One wave owns one 16-row tile; 8 waves per block -> 128 rows.
// grid.x = NBATCH * 32 = 256 blocks of 256 threads.
//
// d2(m,n) = |i_m|^2 + |j_n|^2 - 2<i_m,j_n>. The padded K=4 slot of the
// 16x16x4 fp32 WMMA carries the column-norm term:
//   A'[m][0..2] = -2 * p_i[m],  A'[m][3] = 1
//   B'[0..2][n] =      p_j[n],  B'[3][n] = |p_j[n]|^2
//   C[r]        = |p_i[r]|^2                  (loop-invariant!)
//   D = A' x B' + C  ==>  D[r][n] = d2(r, n)  in ONE instruction per tile.
// The diagonal tile is rotated out of the hot loop (j = (i_tile+jj) & 255,
// jj = 1..255) so the body is branch-free; the self-tile is processed once
// at the end with the self-distance masked.
__global__ __launch_bounds__(256)
void nn_mind2_kernel(const float* __restrict__ centers, float* __restrict__ ws) {
    __shared__ float4 lds4[NPTS * 3 / 4];          // 48 KB: whole batch resident
    float* lds = (float*)lds4;

    const int tid  = threadIdx.x;
    const int wave = tid >> 5;
    const int lane = tid & 31;
    const bool hi  = (lane & 16) != 0;             // false: K={0,1}; true: K={2,3}
    const int l15  = lane & 15;

    const int batch  = blockIdx.x >> 5;
    const int group  = blockIdx.x & 31;
    const int i_tile = group * 8 + wave;           // 0..255
    const int i_base = i_tile * 16;

    // Stage this batch's centers into LDS (float4 loads, 16B-aligned).
    const float4* src = (const float4*)(centers + (size_t)batch * NPTS * 3);
    for (int i = tid; i < NPTS * 3 / 4; i += 256) lds4[i] = src[i];
    __syncthreads();

    // This lane's row point (unconditional 3-component load; operand halves
    // chosen with cndmask — no exec-masked memory ops).
    const int pa = (i_base + l15) * 3;
    const float ax = lds[pa], ay = lds[pa + 1], az = lds[pa + 2];
    const float sq_self = ax * ax + ay * ay + az * az;   // |p(i_base+l15)|^2

    v2f a2;                                        // A' = [-2x,-2y,-2z, 1]
    a2.x = hi ? -2.0f * az : -2.0f * ax;
    a2.y = hi ? 1.0f      : -2.0f * ay;

    // Loop-invariant C operand: the 8 row norms for this lane's C rows
    // (M = r + 8*hi).
    v8f csq;
#pragma unroll
    for (int r = 0; r < 8; ++r)
        csq[r] = __shfl(sq_self, r + (hi ? 8 : 0));

    float mind2[8];
#pragma unroll
    for (int r = 0; r < 8; ++r) mind2[r] = INFINITY;

    // Hot loop: 255 off-diagonal tiles, branch-free body.
    for (int jj = 1; jj < NTILES; ++jj) {
        const int j  = (i_tile + jj) & (NTILES - 1);
        const int pb = (j * 16 + l15) * 3;
        const float bx = lds[pb], by = lds[pb + 1], bz = lds[pb + 2];
        const float sqj = bx * bx + by * by + bz * bz;

        v2f b;                                     // B' = [x', y', z', |p'|^2]
        b.x = hi ? bz  : bx;
        b.y = hi ? sqj : by;

        // c[r] = d2(row r, col l15) straight out of the matrix unit.
        v8f c = __builtin_amdgcn_wmma_f32_16x16x4_f32(
            false, a2, false, b, (short)0, csq, false, false);

#pragma unroll
        for (int r = 0; r < 8; ++r)
            mind2[r] = fminf(mind2[r], c[r]);
    }

    // Self-tile (j == i_tile): B' built from this lane's own point.
    {
        v2f b;
        b.x = hi ? az      : ax;
        b.y = hi ? sq_self : ay;

        v8f c = __builtin_amdgcn_wmma_f32_16x16x4_f32(
            false, a2, false, b, (short)0, csq, false, false);

#pragma unroll
        for (int r = 0; r < 8; ++r) {
            const bool diag = (r + (hi ? 8 : 0)) == l15;
            mind2[r] = fminf(mind2[r], diag ? INFINITY : c[r]);
        }
    }

    // Min over the 16 lanes of each half (columns of each row-tile half).
#pragma unroll
    for (int r = 0; r < 8; ++r) {
#pragma unroll
        for (int m = 1; m <= 8; m <<= 1)
            mind2[r] = fminf(mind2[r], __shfl_xor(mind2[r], m));
    }

    // Lanes 0 and 16 hold finished mins for rows 0-7 / 8-15 of this tile.
    if (l15 == 0) {
        float* dst = ws + (size_t)batch * NPTS + i_base + (hi ? 8 : 0);
#pragma unroll
        for (int r = 0; r < 8; ++r) dst[r] = mind2[r];
    }
}

// Kernel 2: deterministic fixed-order reduction of the 32768 per-row mins.
__global__ __launch_bounds__(256)
void nn_loss_reduce_kernel(const float* __restrict__ ws, float* __restrict__ out) {
    __shared__ float red[256];
    const int total = NBATCH * NPTS;               // 32768
    float s = 0.0f;
    for (int i = threadIdx.x; i < total; i += 256) {
        float d = sqrtf(fmaxf(ws[i], 0.0f)) - TARGET_F;
        s += d * d;
    }
    red[threadIdx.x] = s;
    __syncthreads();
    for (int off = 128; off > 0; off >>= 1) {
        if (threadIdx.x < off) red[threadIdx.x] += red[threadIdx.x + off];
        __syncthreads();
    }
    if (threadIdx.x == 0) out[0] = red[0] * (1.0f / (float)total);
}

extern "C" void kernel_launch(void* const* d_in, const int* in_sizes, int n_in,
                              void* d_out, int out_size, void* d_ws, size_t ws_size,
                              hipStream_t stream) {
    (void)in_sizes; (void)n_in; (void)out_size; (void)ws_size;
    const float* centers = (const float*)d_in[0];
    float* out = (float*)d_out;
    float* ws  = (float*)d_ws;                     // needs 8*4096*4 = 128 KB

    nn_mind2_kernel<<<dim3(NBATCH * 32), dim3(256), 0, stream>>>(centers, ws);
    nn_loss_reduce_kernel<<<dim3(1), dim3(256), 0, stream>>>(ws, out);
}